// HeteroGNN2_76802605187593
// MI455X (gfx1250) — compile-verified
//
#include <hip/hip_runtime.h>
#include <hip/hip_bf16.h>

typedef __attribute__((ext_vector_type(2))) float v2f;
typedef __attribute__((ext_vector_type(8))) float v8f;

#define LDS_STRIDE 132  // 128 + 4 pad: avoids 64-bank conflicts on per-row reads

// ---------------------------------------------------------------------------
// Zero-fill (needed because segment sums accumulate via atomics each call)
// ---------------------------------------------------------------------------
__global__ void hg2_zero_f32(float* __restrict__ p, size_t n) {
    size_t i = (size_t)blockIdx.x * blockDim.x + threadIdx.x;
    size_t stride = (size_t)gridDim.x * blockDim.x;
    for (; i < n; i += stride) p[i] = 0.0f;
}

// ---------------------------------------------------------------------------
// Edge scatter: wave-per-edge. 32 lanes x float4 = 128 features per edge.
// sum[dst] += X[src]; cnt[dst] += 1
// ---------------------------------------------------------------------------
__global__ void hg2_scatter_sum(const float* __restrict__ X,
                                const int* __restrict__ src,
                                const int* __restrict__ dst,
                                float* __restrict__ sum,
                                float* __restrict__ cnt,
                                int E) {
    int gw   = (int)(((size_t)blockIdx.x * blockDim.x + threadIdx.x) >> 5);
    int lane = threadIdx.x & 31;
    int nw   = (int)(((size_t)gridDim.x * blockDim.x) >> 5);
    for (int e = gw; e < E; e += nw) {
        int s = src[e];
        int d = dst[e];
        const float4 v = *reinterpret_cast<const float4*>(X + ((size_t)s << 7) + lane * 4);
        float* p = sum + ((size_t)d << 7) + lane * 4;
        atomicAdd(p + 0, v.x);
        atomicAdd(p + 1, v.y);
        atomicAdd(p + 2, v.z);
        atomicAdd(p + 3, v.w);
        if (lane == 0) atomicAdd(cnt + d, 1.0f);
    }
}

// ---------------------------------------------------------------------------
// Fused SAGE GEMM using V_WMMA_F32_16X16X4_F32 (fp32-exact, wave32):
//   out[N,HOUT] = RELU? ( (A/max(cnt,1)) @ Wl  +  (B @ Wr)  + bias )
//
// Persistent-block design: each block grid-strides over 16-row M-tiles.
// Each wave owns one 16-column N-tile; its Wl/Wr fragments (32 x v2f each)
// are preloaded into registers ONCE and reused for every M-tile, so the
// steady-state inner loop is ds_load_b64 (A frags from LDS) + v_wmma only.
// K = 128 fixed; HOUT is a template constant so all strides are shifts.
//
// Fragment layouts per CDNA5 ISA 7.12.2 (f32):
//   A 16x4 : lanes 0-15 -> M=lane, K={0,1}; lanes 16-31 -> M=lane-16, K={2,3}
//   B 4x16 : lanes 0-15 -> N=lane, K={0,1}; lanes 16-31 -> N=lane-16, K={2,3}
//   C 16x16: vgpr r -> (M=r, N=lane) / (M=r+8, N=lane-16)
// ---------------------------------------------------------------------------
template <int HOUT, bool HASB, bool RELU>
__global__ __launch_bounds__(256)
void hg2_sage_gemm_wmma(const float* __restrict__ A,
                        const float* __restrict__ cnt,
                        const float* __restrict__ B,
                        const float* __restrict__ Wl,
                        const float* __restrict__ Wr,
                        const float* __restrict__ bias,
                        float* __restrict__ out,
                        int N, int numTiles) {
    constexpr int K       = 128;
    constexpr int NTILES  = HOUT / 16;       // waves per block
    constexpr int THREADS = NTILES * 32;     // launch with exactly this many
    constexpr int KSTEPS  = K / 4;           // 32 WMMAs per operand matrix

    __shared__ float sA[16 * LDS_STRIDE];
    __shared__ float sB[HASB ? 16 * LDS_STRIDE : 16];
    __shared__ float sScale[16];

    const int tid  = threadIdx.x;
    const int w    = tid >> 5;
    const int lane = tid & 31;
    const int m    = lane & 15;
    const int koff = (lane >> 4) << 1;       // 0 for lanes 0-15, 2 for lanes 16-31
    const int n    = (w << 4) + m;

    // ---- Preload this wave's weight fragments into registers (once) ------
    v2f fl[KSTEPS];
    #pragma unroll
    for (int j = 0; j < KSTEPS; ++j) {
        int k = (j << 2) + koff;
        fl[j].x = Wl[k * HOUT + n];
        fl[j].y = Wl[(k + 1) * HOUT + n];
    }
    v2f fr[HASB ? KSTEPS : 1];
    if (HASB) {
        #pragma unroll
        for (int j = 0; j < KSTEPS; ++j) {
            int k = (j << 2) + koff;
            fr[j].x = Wr[k * HOUT + n];
            fr[j].y = Wr[(k + 1) * HOUT + n];
        }
    }
    const float bv = bias[n];

    // ---- Grid-stride over M-tiles ---------------------------------------
    for (int t = blockIdx.x; t < numTiles; t += gridDim.x) {
        const int rb = t << 4;

        __syncthreads();  // previous iteration's LDS reads are done
        if (tid < 16) {
            float scl = 1.0f;
            int row = rb + tid;
            if (cnt != nullptr && row < N) scl = 1.0f / fmaxf(cnt[row], 1.0f);
            sScale[tid] = scl;
        }
        __syncthreads();

        // Coalesced staging: 16x128 floats, float4 per thread.
        for (int i = tid * 4; i < 16 * K; i += THREADS * 4) {
            int r = i >> 7;
            int c = i & 127;
            int row = rb + r;
            float4 va = make_float4(0.f, 0.f, 0.f, 0.f);
            float4 vb = make_float4(0.f, 0.f, 0.f, 0.f);
            if (row < N) {
                va = *reinterpret_cast<const float4*>(A + ((size_t)row << 7) + c);
                if (HASB) vb = *reinterpret_cast<const float4*>(B + ((size_t)row << 7) + c);
            }
            float s = sScale[r];
            float* pa = &sA[r * LDS_STRIDE + c];
            pa[0] = va.x * s; pa[1] = va.y * s; pa[2] = va.z * s; pa[3] = va.w * s;
            if (HASB) {
                float* pb = &sB[r * LDS_STRIDE + c];
                pb[0] = vb.x; pb[1] = vb.y; pb[2] = vb.z; pb[3] = vb.w;
            }
        }

        // Speculative prefetch of next tile's rows (global_prefetch_b8).
        {
            int tn = t + gridDim.x;
            if (tn < numTiles && tid < 16) {
                __builtin_prefetch(A + ((size_t)((tn << 4) + tid) << 7), 0, 0);
            }
        }
        __syncthreads();

        // ---- 16x16 tile compute: pure ds_load + wmma chains --------------
        v8f acc = {0.f, 0.f, 0.f, 0.f, 0.f, 0.f, 0.f, 0.f};
        const float* sArow = &sA[m * LDS_STRIDE];
        #pragma unroll
        for (int j = 0; j < KSTEPS; ++j) {
            v2f a;
            a.x = sArow[(j << 2) + koff];
            a.y = sArow[(j << 2) + koff + 1];
            acc = __builtin_amdgcn_wmma_f32_16x16x4_f32(false, a, false, fl[j],
                                                        (short)0, acc, false, false);
        }
        if (HASB) {
            const float* sBrow = &sB[m * LDS_STRIDE];
            #pragma unroll
            for (int j = 0; j < KSTEPS; ++j) {
                v2f a;
                a.x = sBrow[(j << 2) + koff];
                a.y = sBrow[(j << 2) + koff + 1];
                acc = __builtin_amdgcn_wmma_f32_16x16x4_f32(false, a, false, fr[j],
                                                            (short)0, acc, false, false);
            }
        }

        // ---- Epilogue: bias (+ReLU) and store ----------------------------
        const int mbase = rb + ((lane >> 4) << 3);  // lanes>=16 hold rows M=8..15
        #pragma unroll
        for (int r = 0; r < 8; ++r) {
            int row = mbase + r;
            if (row < N) {
                float v = acc[r] + bv;
                if (RELU) v = fmaxf(v, 0.0f);
                out[(size_t)row * HOUT + n] = v;
            }
        }
    }
}

// ---------------------------------------------------------------------------
// Host-side orchestration
// ---------------------------------------------------------------------------
extern "C" void kernel_launch(void* const* d_in, const int* in_sizes, int n_in,
                              void* d_out, int out_size, void* d_ws, size_t ws_size,
                              hipStream_t stream) {
    (void)in_sizes; (void)n_in; (void)out_size;

    const float* x_movie  = (const float*)d_in[0];
    const float* x_user   = (const float*)d_in[1];
    const int*   src_sims = (const int*)d_in[2];
    const int*   dst_sims = (const int*)d_in[3];
    const int*   src_rev  = (const int*)d_in[4];
    const int*   dst_rev  = (const int*)d_in[5];
    const float* W1l = (const float*)d_in[6];
    const float* b1l = (const float*)d_in[7];
    const float* W1r = (const float*)d_in[8];
    const float* W2l = (const float*)d_in[9];
    const float* b2l = (const float*)d_in[10];
    const float* W2r = (const float*)d_in[11];
    const float* W3l = (const float*)d_in[12];
    const float* b3l = (const float*)d_in[13];
    const float* W3r = (const float*)d_in[14];
    const float* Wl1 = (const float*)d_in[15];
    const float* bl1 = (const float*)d_in[16];
    const float* Wl2 = (const float*)d_in[17];
    const float* bl2 = (const float*)d_in[18];
    const float* Wl3 = (const float*)d_in[19];
    const float* bl3 = (const float*)d_in[20];

    float* out = (float*)d_out;
    float* ws  = (float*)d_ws;

    const int NM = 50000, NU = 100000;
    const int ES = 800000, ER = 1600000;
    const size_t NMD = (size_t)NM * 128;   // 6.4M floats
    const size_t NUD = (size_t)NU * 128;   // 12.8M floats

    // Workspace layout (liveness-based reuse):
    //   [0, NUD)          : sum region (sum1 uses first NMD; sum2 / sum3 full)
    //   [NMD, 2*NMD)      : h0 (dead before sum2 is zeroed)
    //   [NUD, NUD+NMD)    : h
    //   [NUD+NMD, ..+NUD) : u0 (later reused as u2)
    //   next NUD          : u
    //   next NM           : cnt1
    //   next NU           : cnt2
    float* sumr = ws;
    float* h0   = ws + NMD;
    float* h    = ws + NUD;
    float* u0   = ws + NUD + NMD;
    float* u    = ws + 2 * NUD + NMD;
    float* u2   = u0;
    float* cnt1 = ws + 3 * NUD + NMD;
    float* cnt2 = cnt1 + NM;

    const size_t need = (3 * NUD + NMD + (size_t)NM + (size_t)NU) * sizeof(float);
    if (ws_size < need) return;

    const int TM = NM / 16;   // 3125 M-tiles
    const int TU = NU / 16;   // 6250 M-tiles
    const int GM = 1024;      // persistent blocks (grid-stride over tiles)
    const int GU = 2048;

    // ---- Layer 1: movie-movie SAGE + dense -------------------------------
    hg2_zero_f32<<<dim3(2048), dim3(256), 0, stream>>>(sumr, NMD);
    hg2_zero_f32<<<dim3(64),   dim3(256), 0, stream>>>(cnt1, (size_t)NM);
    hg2_scatter_sum<<<dim3(4096), dim3(256), 0, stream>>>(x_movie, src_sims, dst_sims,
                                                          sumr, cnt1, ES);
    hg2_sage_gemm_wmma<128, true, true><<<dim3(GM), dim3(256), 0, stream>>>(
        sumr, cnt1, x_movie, W1l, W1r, b1l, h0, NM, TM);
    hg2_sage_gemm_wmma<128, false, true><<<dim3(GM), dim3(256), 0, stream>>>(
        h0, nullptr, nullptr, Wl1, nullptr, bl1, h, NM, TM);

    // ---- Layer 2: movie->user SAGE + dense -------------------------------
    hg2_zero_f32<<<dim3(2048), dim3(256), 0, stream>>>(sumr, NUD);
    hg2_zero_f32<<<dim3(128),  dim3(256), 0, stream>>>(cnt2, (size_t)NU);
    hg2_scatter_sum<<<dim3(8192), dim3(256), 0, stream>>>(x_movie, src_rev, dst_rev,
                                                          sumr, cnt2, ER);
    hg2_sage_gemm_wmma<128, true, true><<<dim3(GU), dim3(256), 0, stream>>>(
        sumr, cnt2, x_user, W2l, W2r, b2l, u0, NU, TU);
    hg2_sage_gemm_wmma<128, false, true><<<dim3(GU), dim3(256), 0, stream>>>(
        u0, nullptr, nullptr, Wl2, nullptr, bl2, u, NU, TU);

    // ---- Layer 3: h(movie)->user SAGE + final projection -----------------
    hg2_zero_f32<<<dim3(2048), dim3(256), 0, stream>>>(sumr, NUD);
    hg2_zero_f32<<<dim3(128),  dim3(256), 0, stream>>>(cnt2, (size_t)NU);
    hg2_scatter_sum<<<dim3(8192), dim3(256), 0, stream>>>(h, src_rev, dst_rev,
                                                          sumr, cnt2, ER);
    hg2_sage_gemm_wmma<128, true, true><<<dim3(GU), dim3(256), 0, stream>>>(
        sumr, cnt2, u, W3l, W3r, b3l, u2, NU, TU);
    // Final projection: HOUT=64 -> 4 waves (128 threads) per block, no ReLU.
    hg2_sage_gemm_wmma<64, false, false><<<dim3(GU), dim3(128), 0, stream>>>(
        u2, nullptr, nullptr, Wl3, nullptr, bl3, out, NU, TU);
}